// KL_Loss_85864986181897
// MI455X (gfx1250) — compile-verified
//
#include <hip/hip_runtime.h>
#include <math.h>

// ---------------------------------------------------------------------------
// KL contrastive loss, MI455X (gfx1250).
//   kld[i,j] = 0.5*(slog[j]-slog[i]-D + P_i·IV_j - 2*MU_i·R_j + c[j])
//   sim = kld/0.1 ; diag = -9e15/0.1 ; loss_i = LSE_j(sim) - sim[i,(i+N)%M]
// Two 4096x4096x256 f32 GEMMs done with V_WMMA_F32_16X16X4_F32 (exact f32
// math — logsumexp at T=0.1 amplifies operand error x10 in the exponent, so
// bf16/f16 WMMA would not validate). Everything L2-resident; compute-bound.
// ---------------------------------------------------------------------------

typedef __attribute__((ext_vector_type(2))) float v2f;
typedef __attribute__((ext_vector_type(4))) float v4f;
typedef __attribute__((ext_vector_type(8))) float v8f;

#define DDIM        256
#define MROWS       4096
#define NHALF       2048
#define ROWS_PER_BLK 32
#define LDS_STRIDE   260   // 260 % 64 == 4 -> conflict-free row gathers

// ---------------------------------------------------------------------------
// Kernel 1: derived operand matrices + per-row reductions (1 wave per row)
// ---------------------------------------------------------------------------
__global__ __launch_bounds__(256) void prep_kernel(
    const float* __restrict__ loc1, const float* __restrict__ sc1,
    const float* __restrict__ loc2, const float* __restrict__ sc2,
    float* __restrict__ MU, float* __restrict__ P,
    float* __restrict__ IV, float* __restrict__ R,
    float* __restrict__ SLOG, float* __restrict__ CV,
    float* __restrict__ out)
{
    if (blockIdx.x == 0 && threadIdx.x == 0) out[0] = 0.0f;  // harness poisons d_out

    const int wave = threadIdx.x >> 5;
    const int lane = threadIdx.x & 31;
    const int row  = blockIdx.x * 8 + wave;

    const float* mu_src  = (row < NHALF) ? (loc1 + (size_t)row * DDIM)
                                         : (loc2 + (size_t)(row - NHALF) * DDIM);
    const float* var_src = (row < NHALF) ? (sc1  + (size_t)row * DDIM)
                                         : (sc2  + (size_t)(row - NHALF) * DDIM);

    const size_t base = (size_t)row * DDIM;
    float slog = 0.0f, cacc = 0.0f;
    for (int d = lane; d < DDIM; d += 32) {
        float m  = mu_src[d];
        float v  = var_src[d];
        float iv = 1.0f / v;
        MU[base + d] = m;
        P [base + d] = m * m + v;
        IV[base + d] = iv;
        R [base + d] = m * iv;
        slog += __logf(v);
        cacc += m * m * iv;
    }
    #pragma unroll
    for (int off = 16; off; off >>= 1) {
        slog += __shfl_xor(slog, off, 32);
        cacc += __shfl_xor(cacc, off, 32);
    }
    if (lane == 0) { SLOG[row] = slog; CV[row] = cacc; }
}

// ---------------------------------------------------------------------------
// Kernel 2: fused WMMA-f32 GEMMs + online logsumexp + mean
// 32 rows per block; 8 waves each own a 16-col tile per 128-col panel.
// ---------------------------------------------------------------------------
__global__ __launch_bounds__(256) void kl_main_kernel(
    const float* __restrict__ MU, const float* __restrict__ P,
    const float* __restrict__ IV, const float* __restrict__ R,
    const float* __restrict__ SLOG, const float* __restrict__ CV,
    float* __restrict__ out)
{
    __shared__ __align__(16) float lP[ROWS_PER_BLK * LDS_STRIDE];
    __shared__ __align__(16) float lM[ROWS_PER_BLK * LDS_STRIDE];
    __shared__ float partMax[8][ROWS_PER_BLK];
    __shared__ float partSum[8][ROWS_PER_BLK];
    __shared__ float partPos[8][ROWS_PER_BLK];

    const int tid  = threadIdx.x;
    const int wave = tid >> 5;
    const int lane = tid & 31;
    const int half = lane >> 4;     // which 16-lane group
    const int ln16 = lane & 15;
    const int rb   = blockIdx.x * ROWS_PER_BLK;

    // Stage this block's 32 rows of P and MU into padded LDS (coalesced b128).
    for (int idx = tid; idx < ROWS_PER_BLK * (DDIM / 4); idx += 256) {
        int r = idx >> 6;   // 64 float4 segments per row
        int s = idx & 63;
        v4f pv = *(const v4f*)(P  + (size_t)(rb + r) * DDIM + s * 4);
        v4f mv = *(const v4f*)(MU + (size_t)(rb + r) * DDIM + s * 4);
        *(v4f*)(&lP[r * LDS_STRIDE + s * 4]) = pv;
        *(v4f*)(&lM[r * LDS_STRIDE + s * 4]) = mv;
    }
    __syncthreads();

    // Row-side scalar term: 5*slog[i] for this lane's 16 (tile,reg) rows.
    float rowterm[2][8];
    #pragma unroll
    for (int t = 0; t < 2; ++t)
        #pragma unroll
        for (int r = 0; r < 8; ++r)
            rowterm[t][r] = 5.0f * SLOG[rb + t * 16 + r + 8 * half];

    float runMax[2][8], runSum[2][8], posv[2][8];
    #pragma unroll
    for (int t = 0; t < 2; ++t)
        #pragma unroll
        for (int r = 0; r < 8; ++r) {
            runMax[t][r] = -__builtin_inff();
            runSum[t][r] = 0.0f;
            posv  [t][r] = -__builtin_inff();
        }

    const int aoff0 = (0 * 16 + ln16) * LDS_STRIDE + 2 * half;
    const int aoff1 = (1 * 16 + ln16) * LDS_STRIDE + 2 * half;

    for (int panel = 0; panel < MROWS / 128; ++panel) {
        const int cb = panel * 128 + wave * 16;       // this wave's column tile
        const int j  = cb + ln16;                     // this lane's column
        const float colterm = 5.0f * (SLOG[j] + CV[j]);

        // B fragments gathered straight from L2-resident global (b64/lane).
        const float* ivb = IV + (size_t)j * DDIM + 2 * half;
        const float* rbp = R  + (size_t)j * DDIM + 2 * half;

        v8f accA0 = {}, accA1 = {}, accB0 = {}, accB1 = {};
        #pragma unroll 8
        for (int kb = 0; kb < DDIM; kb += 4) {
            v2f bIV = *(const v2f*)(ivb + kb);
            v2f bR  = *(const v2f*)(rbp + kb);
            v2f aP0 = *(const v2f*)(&lP[aoff0 + kb]);
            v2f aM0 = *(const v2f*)(&lM[aoff0 + kb]);
            v2f aP1 = *(const v2f*)(&lP[aoff1 + kb]);
            v2f aM1 = *(const v2f*)(&lM[aoff1 + kb]);
            accA0 = __builtin_amdgcn_wmma_f32_16x16x4_f32(false, aP0, false, bIV, (short)0, accA0, false, false);
            accB0 = __builtin_amdgcn_wmma_f32_16x16x4_f32(false, aM0, false, bR,  (short)0, accB0, false, false);
            accA1 = __builtin_amdgcn_wmma_f32_16x16x4_f32(false, aP1, false, bIV, (short)0, accA1, false, false);
            accB1 = __builtin_amdgcn_wmma_f32_16x16x4_f32(false, aM1, false, bR,  (short)0, accB1, false, false);
        }

        // Fused epilogue: scalar terms, diag fill, pos capture, online LSE.
        #pragma unroll
        for (int t = 0; t < 2; ++t) {
            #pragma unroll
            for (int r = 0; r < 8; ++r) {
                float a = (t == 0) ? accA0[r] : accA1[r];
                float b = (t == 0) ? accB0[r] : accB1[r];
                int   i = rb + t * 16 + r + 8 * half;
                float sim = 5.0f * a - 10.0f * b + colterm - rowterm[t][r]
                            - 5.0f * (float)DDIM;
                if (j == i) sim = -9.0e16f;                       // NEG_FILL / T
                if (j == ((i + NHALF) & (MROWS - 1))) posv[t][r] = sim;
                float nm = fmaxf(runMax[t][r], sim);
                runSum[t][r] = runSum[t][r] * __expf(runMax[t][r] - nm)
                               + __expf(sim - nm);
                runMax[t][r] = nm;
            }
        }
    }

    // Reduce LSE across the 16 lanes that share a row, then stash per-wave.
    #pragma unroll
    for (int t = 0; t < 2; ++t)
        #pragma unroll
        for (int r = 0; r < 8; ++r) {
            float m = runMax[t][r], s = runSum[t][r], p = posv[t][r];
            #pragma unroll
            for (int off = 8; off; off >>= 1) {
                float om = __shfl_xor(m, off, 32);
                float os = __shfl_xor(s, off, 32);
                float op = __shfl_xor(p, off, 32);
                float nm = fmaxf(m, om);
                s = s * __expf(m - nm) + os * __expf(om - nm);
                m = nm;
                p = fmaxf(p, op);
            }
            if (ln16 == 0) {
                int rl = t * 16 + r + 8 * half;
                partMax[wave][rl] = m;
                partSum[wave][rl] = s;
                partPos[wave][rl] = p;
            }
        }
    __syncthreads();

    // One thread per row merges the 8 wave partials and accumulates the mean.
    if (tid < ROWS_PER_BLK) {
        float m = -__builtin_inff(), s = 0.0f, p = -__builtin_inff();
        #pragma unroll
        for (int w = 0; w < 8; ++w) {
            float om = partMax[w][tid], os = partSum[w][tid], op = partPos[w][tid];
            float nm = fmaxf(m, om);
            s = s * __expf(m - nm) + os * __expf(om - nm);
            m = nm;
            p = fmaxf(p, op);
        }
        float loss = m + __logf(s) - p;
        atomicAdd(out, loss * (1.0f / (float)MROWS));
    }
}

// ---------------------------------------------------------------------------
extern "C" void kernel_launch(void* const* d_in, const int* in_sizes, int n_in,
                              void* d_out, int out_size, void* d_ws, size_t ws_size,
                              hipStream_t stream)
{
    const float* loc1 = (const float*)d_in[0];
    const float* sc1  = (const float*)d_in[1];
    const float* loc2 = (const float*)d_in[2];
    const float* sc2  = (const float*)d_in[3];

    float* ws   = (float*)d_ws;                 // ~16.8 MB of scratch used
    float* MU   = ws;
    float* P    = MU + (size_t)MROWS * DDIM;
    float* IV   = P  + (size_t)MROWS * DDIM;
    float* R    = IV + (size_t)MROWS * DDIM;
    float* SLOG = R  + (size_t)MROWS * DDIM;
    float* CV   = SLOG + MROWS;
    float* out  = (float*)d_out;

    prep_kernel<<<MROWS / 8, 256, 0, stream>>>(loc1, sc1, loc2, sc2,
                                               MU, P, IV, R, SLOG, CV, out);
    kl_main_kernel<<<MROWS / ROWS_PER_BLK, 256, 0, stream>>>(MU, P, IV, R,
                                                             SLOG, CV, out);
}